// SS2D_55190329754138
// MI455X (gfx1250) — compile-verified
//
#include <hip/hip_runtime.h>

// SS2D selective scan for MI455X (gfx1250, wave32).
// 12 waves/block; wave w computes output-channel tile [16w,16w+16) of the
// 16-sequence state update H_t = H_{t-1} @ A + Bv*x_t via V_WMMA_F32_16X16X4_F32.

#define DIMC   192     // channels
#define KSTEPS 48      // 192 / 4 (WMMA K)
#define HPAD   196     // padded LDS row stride (floats) -> conflict-free
#define LSTEPS 128     // scan length

typedef __attribute__((ext_vector_type(2))) float v2f;
typedef __attribute__((ext_vector_type(8))) float v8f;

__global__ __launch_bounds__(384) void ss2d_scan(
    const float* __restrict__ x, const float* __restrict__ A,
    const float* __restrict__ Bv, const float* __restrict__ Cv,
    const float* __restrict__ Dv, const float* __restrict__ res,
    float* __restrict__ out, int dir_base, int accumulate)
{
    __shared__ __align__(16) float Hs[16 * HPAD];

    const int tid  = threadIdx.x;
    const int wv   = tid >> 5;      // wave id 0..11 -> channel tile
    const int lane = tid & 31;
    const int half = lane >> 4;     // 0: lanes 0-15, 1: lanes 16-31
    const int n    = lane & 15;
    const int ch   = (wv << 4) + n; // this lane's output channel (D-tile column)

    int dir, g;
    if (dir_base >= 2) { dir = dir_base + (int)blockIdx.x; g = 0; }
    else               { dir = dir_base; g = (int)blockIdx.x; }

    const float bv = Bv[ch];
    const float cv = Cv[ch];
    const float dr = Dv[ch] + res[0];

    // ---- A fragments: WMMA B operand (K rows 4k..4k+3, cols 16w..16w+15) ----
    // B layout (4x16 f32): lanes 0-15 hold N, vgpr0=K0,vgpr1=K1; lanes 16-31 K2,K3.
    v2f afrag[KSTEPS];
    {
        const float* ap = A + ch;
        #pragma unroll
        for (int k = 0; k < KSTEPS; ++k) {
            const int r = 4 * k + 2 * half;
            afrag[k].x = ap[r * DIMC];
            afrag[k].y = ap[(r + 1) * DIMC];
        }
    }

    // ---- per-sequence element offsets (this lane touches seqs v+8*half) ----
    unsigned off[8];
    float    msk[8];
    #pragma unroll
    for (int v = 0; v < 8; ++v) {
        const int s = v + (half << 3);
        unsigned base; bool ok = true;
        if (dir == 0) {            // horizontal: x[b, t, w0+s, c]
            const int b = g >> 3, w0 = (g & 7) << 4;
            base = (unsigned)(b * 16384 + (w0 + s)) * DIMC;
        } else if (dir == 1) {     // vertical: x[b, h0+s, t, c]
            const int b = g >> 3, h0 = (g & 7) << 4;
            base = (unsigned)(b * 16384 + (h0 + s) * 128) * DIMC;
        } else if (dir == 2) {     // diagonal: x[s, t, t, c]
            base = (unsigned)(s * 16384) * DIMC;        ok = (s < 8);
        } else {                   // anti-diagonal: x[s, t, 127-t, c]
            base = (unsigned)(s * 16384 + 127) * DIMC;  ok = (s < 8);
        }
        off[v] = (ok ? base : 0u) + (unsigned)ch;
        msk[v] = ok ? 1.0f : 0.0f;
    }
    const unsigned stride = (dir == 0) ? 128u * DIMC
                          : (dir == 1) ? (unsigned)DIMC
                          : (dir == 2) ? 129u * DIMC
                                       : 127u * DIMC;

    // ---- zero initial state ----
    for (int i = tid; i < 16 * HPAD; i += 384) Hs[i] = 0.0f;
    __syncthreads();

    // A-operand (16x4 f32) read base: lane<16 -> H[seq=n][4k..4k+1],
    // lane>=16 -> H[seq=n][4k+2..4k+3]. 8B aligned (HPAD even).
    const float* Hrd = &Hs[n * HPAD + 2 * half];
    // D-tile store base: row (v + 8*half), col 16w+n.
    float*       Hwr = &Hs[(half << 3) * HPAD + (wv << 4) + n];

    #pragma unroll 1
    for (int t = 0; t < LSTEPS; ++t) {
        // x_t tile (masked for padded diag sequences)
        float xv[8];
        #pragma unroll
        for (int v = 0; v < 8; ++v) xv[v] = x[off[v]] * msk[v];

        // acc = H_{t-1} @ A  (this wave's 16 columns); two accumulators for ILP
        v8f acc0 = {0.f,0.f,0.f,0.f,0.f,0.f,0.f,0.f};
        v8f acc1 = acc0;
        #pragma unroll
        for (int k = 0; k < KSTEPS; k += 2) {
            const v2f h0 = *(const v2f*)(Hrd + 4 * k);
            acc0 = __builtin_amdgcn_wmma_f32_16x16x4_f32(
                false, h0, false, afrag[k],     (short)0, acc0, false, false);
            const v2f h1 = *(const v2f*)(Hrd + 4 * (k + 1));
            acc1 = __builtin_amdgcn_wmma_f32_16x16x4_f32(
                false, h1, false, afrag[k + 1], (short)0, acc1, false, false);
        }

        // H_t = acc + Bv*x_t ; y_t = Cv*H_t + (Dv+res)*x_t
        float hnew[8];
        #pragma unroll
        for (int v = 0; v < 8; ++v)
            hnew[v] = acc0[v] + acc1[v] + bv * xv[v];

        #pragma unroll
        for (int v = 0; v < 8; ++v) {
            const float y = cv * hnew[v] + dr * xv[v];
            if (msk[v] != 0.0f) {
                if (accumulate) out[off[v]] += y;   // element-disjoint RMW
                else            out[off[v]]  = y;
            }
        }

        __syncthreads();                 // all waves done reading H_{t-1}
        #pragma unroll
        for (int v = 0; v < 8; ++v) Hwr[v * HPAD] = hnew[v];
        __syncthreads();                 // H_t visible to all waves

        #pragma unroll
        for (int v = 0; v < 8; ++v) off[v] += stride;
    }
}

extern "C" void kernel_launch(void* const* d_in, const int* in_sizes, int n_in,
                              void* d_out, int out_size, void* d_ws, size_t ws_size,
                              hipStream_t stream) {
    (void)in_sizes; (void)n_in; (void)out_size; (void)d_ws; (void)ws_size;
    const float* x   = (const float*)d_in[0];
    const float* A   = (const float*)d_in[1];
    const float* Bv  = (const float*)d_in[2];
    const float* Cv  = (const float*)d_in[3];
    const float* Dv  = (const float*)d_in[4];
    const float* res = (const float*)d_in[5];
    float* out = (float*)d_out;

    dim3 blk(384);
    // horizontal: dense write of every output element (no pre-zero needed)
    ss2d_scan<<<64, blk, 0, stream>>>(x, A, Bv, Cv, Dv, res, out, 0, 0);
    // vertical: accumulate (stream-ordered after horizontal)
    ss2d_scan<<<64, blk, 0, stream>>>(x, A, Bv, Cv, Dv, res, out, 1, 1);
    // diagonal + anti-diagonal (mutually disjoint element sets): accumulate
    ss2d_scan<<<2,  blk, 0, stream>>>(x, A, Bv, Cv, Dv, res, out, 2, 1);
}